// Codec_59760174956545
// MI455X (gfx1250) — compile-verified
//
#include <hip/hip_runtime.h>
#include <math.h>

typedef float v2f __attribute__((ext_vector_type(2)));
typedef float v8f __attribute__((ext_vector_type(8)));

#define HH 512
#define WW 512
#define NIMG 12
#define NEG_SLOPE 0.01f
#define LDSW (WW + 8)   // 512 + 6 halo + pad

struct NetW {
    const float *wT, *bT, *wL, *w1, *b1, *w2, *b2, *w3, *b3;
};

__device__ __forceinline__ float ctxW(const float* wT, const float* wL, int ch, int k) {
    // Fused context weight matrix [32 x 24]: taps 0..20 = 3x7 top kernel, 21..23 = 1x3 left kernel
    return (k < 21) ? wT[ch * 21 + k] : wL[ch * 3 + (k - 21)];
}

__device__ __forceinline__ v8f lrelu8(v8f v) {
#pragma unroll
    for (int i = 0; i < 8; ++i) v[i] = v[i] > 0.f ? v[i] : NEG_SLOPE * v[i];
    return v;
}

// Run one causal network for a 16-pixel tile. bfrag = layer-0 B fragments (context),
// st = this wave's private LDS staging [32*16]. Returns clipped prediction (valid in all lanes).
__device__ __forceinline__ float run_net(const NetW& net, const v2f bfrag[6],
                                         float* st, int n, int half) {
    // ---- layer 0: 32x24 context GEMM, K = 24 = 6 chained f32 WMMAs per M-block ----
    v8f d0[2];
#pragma unroll
    for (int mb = 0; mb < 2; ++mb) {
        v8f c;
#pragma unroll
        for (int r = 0; r < 8; ++r) c[r] = net.bT[mb * 16 + half * 8 + r];
        const int ch = mb * 16 + n;  // A-matrix M index = lane%16
#pragma unroll
        for (int kk = 0; kk < 6; ++kk) {
            const int k0 = kk * 4 + half * 2;
            v2f a;
            a.x = ctxW(net.wT, net.wL, ch, k0);
            a.y = ctxW(net.wT, net.wL, ch, k0 + 1);
            c = __builtin_amdgcn_wmma_f32_16x16x4_f32(false, a, false, bfrag[kk],
                                                      (short)0, c, false, false);
        }
        d0[mb] = lrelu8(c);
    }
    // D-layout -> LDS [ch][px]
#pragma unroll
    for (int mb = 0; mb < 2; ++mb)
#pragma unroll
        for (int r = 0; r < 8; ++r) st[(mb * 16 + half * 8 + r) * 16 + n] = d0[mb][r];
    __asm volatile("s_wait_dscnt 0" ::: "memory");

    // ---- layer 1: 32x32 GEMM (K = 32 = 8 WMMAs / M-block) ----
    v2f bf[8];
#pragma unroll
    for (int kk = 0; kk < 8; ++kk) {
        const int k0 = kk * 4 + half * 2;
        bf[kk].x = st[k0 * 16 + n];
        bf[kk].y = st[(k0 + 1) * 16 + n];
    }
    v8f d1[2];
#pragma unroll
    for (int mb = 0; mb < 2; ++mb) {
        v8f c;
#pragma unroll
        for (int r = 0; r < 8; ++r) c[r] = net.b1[mb * 16 + half * 8 + r];
        const int ch = mb * 16 + n;
#pragma unroll
        for (int kk = 0; kk < 8; ++kk) {
            const int k0 = kk * 4 + half * 2;
            v2f a;
            a.x = net.w1[ch * 32 + k0];
            a.y = net.w1[ch * 32 + k0 + 1];
            c = __builtin_amdgcn_wmma_f32_16x16x4_f32(false, a, false, bf[kk],
                                                      (short)0, c, false, false);
        }
        d1[mb] = lrelu8(c);
    }
#pragma unroll
    for (int mb = 0; mb < 2; ++mb)
#pragma unroll
        for (int r = 0; r < 8; ++r) st[(mb * 16 + half * 8 + r) * 16 + n] = d1[mb][r];
    __asm volatile("s_wait_dscnt 0" ::: "memory");

    // ---- layer 2: 32x32 GEMM ----
#pragma unroll
    for (int kk = 0; kk < 8; ++kk) {
        const int k0 = kk * 4 + half * 2;
        bf[kk].x = st[k0 * 16 + n];
        bf[kk].y = st[(k0 + 1) * 16 + n];
    }
    v8f d2[2];
#pragma unroll
    for (int mb = 0; mb < 2; ++mb) {
        v8f c;
#pragma unroll
        for (int r = 0; r < 8; ++r) c[r] = net.b2[mb * 16 + half * 8 + r];
        const int ch = mb * 16 + n;
#pragma unroll
        for (int kk = 0; kk < 8; ++kk) {
            const int k0 = kk * 4 + half * 2;
            v2f a;
            a.x = net.w2[ch * 32 + k0];
            a.y = net.w2[ch * 32 + k0 + 1];
            c = __builtin_amdgcn_wmma_f32_16x16x4_f32(false, a, false, bf[kk],
                                                      (short)0, c, false, false);
        }
        d2[mb] = lrelu8(c);
    }

    // ---- layer 3: 32 -> 1 projection, straight from D-layout registers ----
    float part = 0.f;
#pragma unroll
    for (int r = 0; r < 8; ++r) {
        part += net.w3[half * 8 + r]      * d2[0][r];
        part += net.w3[16 + half * 8 + r] * d2[1][r];
    }
    part += __shfl_xor(part, 16);      // combine the two lane-halves (channels split over lanes)
    float p = part + net.b3[0];
    return fminf(1.f, fmaxf(-1.f, p));
}

__global__ __launch_bounds__(256) void codec_main(
    const float* __restrict__ x, NetW na, NetW nb, NetW nc,
    float* __restrict__ sumsq, int* __restrict__ hist0, int* __restrict__ histd) {
    __shared__ float lds_x[4][LDSW];      // rows h-3..h, col c holds x col (c-3), zero-padded
    __shared__ float stage[8][32 * 16];   // per-wave activation staging
    __shared__ int lh0[256], lhd[256];

    const int img = blockIdx.y;
    const int h = blockIdx.x;
    const int tid = threadIdx.x;

    if (tid < 256) { lh0[tid] = 0; lhd[tid] = 0; }

    const float* ximg = x + (size_t)img * HH * WW;
    for (int i = tid; i < 4 * (WW + 6); i += 256) {
        const int r = i / (WW + 6), c = i % (WW + 6);
        const int xr = h - 3 + r, xc = c - 3;
        float v = 0.f;
        if (xr >= 0 && xc >= 0 && xc < WW) v = ximg[xr * WW + xc];
        lds_x[r][c] = v;
    }
    __syncthreads();

    const int wave = tid >> 5, lane = tid & 31;
    const int n = lane & 15, half = lane >> 4;
    float* st = stage[wave];

    float acc_d = 0.f, acc_x = 0.f;

#pragma unroll 1
    for (int t = 0; t < 4; ++t) {
        const int w0 = wave * 64 + t * 16;  // tile's first column

        // Layer-0 B fragments: context matrix ctx[24 x 16px] from the LDS halo tile.
        v2f bfrag[6];
#pragma unroll
        for (int kk = 0; kk < 6; ++kk) {
            const int k0 = kk * 4 + half * 2;
#pragma unroll
            for (int v = 0; v < 2; ++v) {
                const int k = k0 + v;
                float val;
                if (k < 21) val = lds_x[k / 7][w0 + n + (k % 7)];
                else        val = lds_x[3][w0 + n + (k - 21)];
                if (v == 0) bfrag[kk].x = val; else bfrag[kk].y = val;
            }
        }

        const float pa = run_net(na, bfrag, st, n, half);
        const float pb = run_net(nb, bfrag, st, n, half);
        const float pc = run_net(nc, bfrag, st, n, half);

        if (half == 0) {  // lanes 0..15 own pixels w0+n
            const float xv = lds_x[3][w0 + n + 3];
            const float med = fmaxf(fminf(pa, fmaxf(pb, pc)), fminf(pb, pc));
            const float dv = fmodf(xv - med + 1.f, 2.f) - 1.f;
            acc_d += dv * dv;
            acc_x += xv * xv;
            if (xv >= -1.f && xv <= 1.f) {
                int b = min(max((int)floorf((xv + 1.f) * 128.f), 0), 255);
                atomicAdd(&lh0[b], 1);
            }
            if (dv >= -1.f && dv <= 1.f) {
                int b = min(max((int)floorf((dv + 1.f) * 128.f), 0), 255);
                atomicAdd(&lhd[b], 1);
            }
        }
    }

    // cross-lane sum-of-squares reduction, one global atomic per wave
#pragma unroll
    for (int off = 16; off > 0; off >>= 1) {
        acc_d += __shfl_xor(acc_d, off);
        acc_x += __shfl_xor(acc_x, off);
    }
    if (lane == 0) {
        atomicAdd(&sumsq[0], acc_d);
        atomicAdd(&sumsq[1], acc_x);
    }

    __syncthreads();
    if (tid < 256) {
        if (lh0[tid]) atomicAdd(&hist0[img * 256 + tid], lh0[tid]);
        if (lhd[tid]) atomicAdd(&histd[img * 256 + tid], lhd[tid]);
    }
}

__global__ void codec_zero(float* sumsq, int* h0, int* hd) {
    const int i = blockIdx.x * blockDim.x + threadIdx.x;
    if (i < 2) sumsq[i] = 0.f;
    if (i < NIMG * 256) { h0[i] = 0; hd[i] = 0; }
}

__global__ __launch_bounds__(256) void codec_reduce(
    const float* __restrict__ sumsq, const int* __restrict__ hist0,
    const int* __restrict__ histd, float* __restrict__ out) {
    __shared__ float red[256];
    const int tid = threadIdx.x;
    const float res = (float)(HH * WW);
    float e0 = 0.f, e1 = 0.f;
    for (int i = tid; i < NIMG * 256; i += 256) {
        const float p0 = (float)hist0[i] / res;
        if (p0 > 0.f) e0 -= p0 * log2f(p0);
        const float p1 = (float)histd[i] / res;
        if (p1 > 0.f) e1 -= p1 * log2f(p1);
    }
    red[tid] = e0; __syncthreads();
    for (int s = 128; s > 0; s >>= 1) { if (tid < s) red[tid] += red[tid + s]; __syncthreads(); }
    const float E0 = red[0]; __syncthreads();
    red[tid] = e1; __syncthreads();
    for (int s = 128; s > 0; s >>= 1) { if (tid < s) red[tid] += red[tid + s]; __syncthreads(); }
    const float E1 = red[0];
    if (tid == 0) {
        const float total = (float)NIMG * HH * WW;
        out[0] = 255.f * sqrtf(sumsq[0] / total);   // loss1 (deltas RMSE)
        out[1] = 255.f * sqrtf(sumsq[1] / total);   // loss0 (x RMSE)
        out[2] = E0 / (8.f * NIMG);                 // invCR0
        out[3] = E1 / (8.f * NIMG);                 // invCR1
    }
}

extern "C" void kernel_launch(void* const* d_in, const int* in_sizes, int n_in,
                              void* d_out, int out_size, void* d_ws, size_t ws_size,
                              hipStream_t stream) {
    const float* x = (const float*)d_in[0];
    NetW nets[3];
    for (int p = 0; p < 3; ++p) {
        const int b = 1 + 9 * p;
        nets[p].wT = (const float*)d_in[b + 0];
        nets[p].bT = (const float*)d_in[b + 1];
        nets[p].wL = (const float*)d_in[b + 2];
        nets[p].w1 = (const float*)d_in[b + 3];
        nets[p].b1 = (const float*)d_in[b + 4];
        nets[p].w2 = (const float*)d_in[b + 5];
        nets[p].b2 = (const float*)d_in[b + 6];
        nets[p].w3 = (const float*)d_in[b + 7];
        nets[p].b3 = (const float*)d_in[b + 8];
    }
    float* sumsq = (float*)d_ws;
    int* hist0 = (int*)((char*)d_ws + 16);
    int* histd = hist0 + NIMG * 256;

    codec_zero<<<NIMG, 256, 0, stream>>>(sumsq, hist0, histd);
    codec_main<<<dim3(HH, NIMG), 256, 0, stream>>>(x, nets[0], nets[1], nets[2],
                                                   sumsq, hist0, histd);
    codec_reduce<<<1, 256, 0, stream>>>(sumsq, hist0, histd, (float*)d_out);
}